// embedding_pooling_38878043963634
// MI455X (gfx1250) — compile-verified
//
#include <hip/hip_runtime.h>
#include <float.h>

typedef float v4f __attribute__((ext_vector_type(4)));
typedef int   v4i __attribute__((ext_vector_type(4)));

#define NCLS 5
#define LLEN 512
#define DDIM 256
#define NTHREADS 512
#define LCHUNKS 8          // L split across 8 wave-pairs
#define ROWS 64            // rows per thread = LLEN / LCHUNKS

__device__ __forceinline__ void vmax4(v4f& a, const v4f b) {
  a.x = fmaxf(a.x, b.x);
  a.y = fmaxf(a.y, b.y);
  a.z = fmaxf(a.z, b.z);
  a.w = fmaxf(a.w, b.w);
}

// c is wave-uniform (SGPR via readfirstlane): scalar compare + branch, only
// the taken class pays 4 v_max ops.
__device__ __forceinline__ void update(int c, const v4f v,
    v4f& a0, v4f& a1, v4f& a2, v4f& a3, v4f& a4) {
  if (c == 1)      vmax4(a0, v);
  else if (c == 2) vmax4(a1, v);
  else if (c == 3) vmax4(a2, v);
  else if (c == 4) vmax4(a3, v);
  else if (c == 5) vmax4(a4, v);
}

__global__ __launch_bounds__(NTHREADS) void masked_maxpool_kernel(
    const float* __restrict__ x, const int* __restrict__ labels,
    float* __restrict__ out) {
  __shared__ __align__(16) int s_lab[LLEN];
  __shared__ v4f s_red[LCHUNKS][NCLS][DDIM / 4];   // 40 KB partial maxima

  const int tid    = threadIdx.x;
  const int b      = blockIdx.x;
  const int d4     = tid & 63;   // which float4 of D
  const int lchunk = tid >> 6;   // which 1/8 of L (wave-uniform)

  // Stage this batch's 512 labels into LDS (1 load per thread).
  s_lab[tid] = labels[b * LLEN + tid];
  __syncthreads();

  const float NEG = -FLT_MAX;           // == jnp.finfo(f32).min
  v4f a0 = {NEG, NEG, NEG, NEG};
  v4f a1 = a0, a2 = a0, a3 = a0, a4 = a0;

  const float* xp = x + (size_t)b * LLEN * DDIM
                      + (size_t)(lchunk * ROWS) * DDIM + d4 * 4;
  const int lbase = lchunk * ROWS;

  // Stream 64 rows, unrolled x8: eight global_load_b128 (RT hint: the 128 MB
  // working set is L2-resident across graph replays) in flight per iteration.
  for (int i = 0; i < ROWS; i += 8) {
    __builtin_prefetch(xp + (size_t)(i + 16) * DDIM, 0, 0); // global_prefetch_b8
    v4f v0 = *(const v4f*)(xp + (size_t)(i + 0) * DDIM);
    v4f v1 = *(const v4f*)(xp + (size_t)(i + 1) * DDIM);
    v4f v2 = *(const v4f*)(xp + (size_t)(i + 2) * DDIM);
    v4f v3 = *(const v4f*)(xp + (size_t)(i + 3) * DDIM);
    v4f v4 = *(const v4f*)(xp + (size_t)(i + 4) * DDIM);
    v4f v5 = *(const v4f*)(xp + (size_t)(i + 5) * DDIM);
    v4f v6 = *(const v4f*)(xp + (size_t)(i + 6) * DDIM);
    v4f v7 = *(const v4f*)(xp + (size_t)(i + 7) * DDIM);
    v4i labA = *(const v4i*)&s_lab[lbase + i];
    v4i labB = *(const v4i*)&s_lab[lbase + i + 4];
    // Labels are wave-uniform (all lanes share the same l) -> force SGPR.
    int c0 = __builtin_amdgcn_readfirstlane(labA.x);
    int c1 = __builtin_amdgcn_readfirstlane(labA.y);
    int c2 = __builtin_amdgcn_readfirstlane(labA.z);
    int c3 = __builtin_amdgcn_readfirstlane(labA.w);
    int c4 = __builtin_amdgcn_readfirstlane(labB.x);
    int c5 = __builtin_amdgcn_readfirstlane(labB.y);
    int c6 = __builtin_amdgcn_readfirstlane(labB.z);
    int c7 = __builtin_amdgcn_readfirstlane(labB.w);
    update(c0, v0, a0, a1, a2, a3, a4);
    update(c1, v1, a0, a1, a2, a3, a4);
    update(c2, v2, a0, a1, a2, a3, a4);
    update(c3, v3, a0, a1, a2, a3, a4);
    update(c4, v4, a0, a1, a2, a3, a4);
    update(c5, v5, a0, a1, a2, a3, a4);
    update(c6, v6, a0, a1, a2, a3, a4);
    update(c7, v7, a0, a1, a2, a3, a4);
  }

  // Reduce the 8 L-chunks through LDS.
  s_red[lchunk][0][d4] = a0;
  s_red[lchunk][1][d4] = a1;
  s_red[lchunk][2][d4] = a2;
  s_red[lchunk][3][d4] = a3;
  s_red[lchunk][4][d4] = a4;
  __syncthreads();

  if (tid < NCLS * (DDIM / 4)) {                 // 320 jobs
    const int c  = tid >> 6;
    const int dd = tid & 63;
    v4f m = s_red[0][c][dd];
    vmax4(m, s_red[1][c][dd]);
    vmax4(m, s_red[2][c][dd]);
    vmax4(m, s_red[3][c][dd]);
    vmax4(m, s_red[4][c][dd]);
    vmax4(m, s_red[5][c][dd]);
    vmax4(m, s_red[6][c][dd]);
    vmax4(m, s_red[7][c][dd]);
    // ReLU also realizes the "absent class -> 0" rule (-FLT_MAX -> 0).
    m.x = fmaxf(m.x, 0.0f);
    m.y = fmaxf(m.y, 0.0f);
    m.z = fmaxf(m.z, 0.0f);
    m.w = fmaxf(m.w, 0.0f);
    *(v4f*)(out + (size_t)b * (NCLS * DDIM) + c * DDIM + dd * 4) = m;
  }
}

extern "C" void kernel_launch(void* const* d_in, const int* in_sizes, int n_in,
                              void* d_out, int out_size, void* d_ws, size_t ws_size,
                              hipStream_t stream) {
  const float* x      = (const float*)d_in[0];   // (256, 512, 256) f32
  const int*   labels = (const int*)d_in[1];     // (256, 512, 1) int
  float*       out    = (float*)d_out;           // (256, 1280) f32
  (void)in_sizes; (void)n_in; (void)out_size; (void)d_ws; (void)ws_size;
  masked_maxpool_kernel<<<256, NTHREADS, 0, stream>>>(x, labels, out);
}